// GNNLocalCluster0_F_6158983102548
// MI455X (gfx1250) — compile-verified
//
#include <hip/hip_runtime.h>
#include <hip/hip_bf16.h>

typedef __attribute__((ext_vector_type(2))) float v2f;
typedef __attribute__((ext_vector_type(8))) float v8f;

#define BATCH  4
#define CCH    128
#define HDIM   112
#define HWSZ   (112*112)
#define NPATCH 196      // 4 * 7 * 7
#define NNODE  256      // 16 * 16 nodes per patch
#define CF     16       // projected feature channels
#define DAUG   18       // CF + 2 grid coords
#define DPAD   20       // padded to multiple of 4 for K-steps
#define KTOP   15

__device__ __forceinline__ float grid_norm(int i) {
    // mean 7.5 ; std(ddof=1) of {0..15 each x16} = sqrt(16*340/255)
    const float gstd = 4.618802153517006f;
    return ((float)i - 7.5f) / (gstd + 1e-5f);
}

// ---------------------------------------------------------------------------
// Kernel 1: 1x1 conv (f32 WMMA), grid augment, L2-normalize -> xn[P][256][20]
// ---------------------------------------------------------------------------
__global__ void k1_conv_feat(const float* __restrict__ x,
                             const float* __restrict__ fw,
                             const float* __restrict__ fb,
                             float* __restrict__ xn) {
    __shared__ float fw_s[CF * CCH];
    __shared__ float fb_s[CF];
    __shared__ float ftile[NNODE * CF];

    const int tid = threadIdx.x;
    const int p   = blockIdx.x;
    const int b   = p / 49;
    const int pr  = p % 49;
    const int wgi = pr / 7;
    const int hgj = pr % 7;

    for (int i = tid; i < CF * CCH; i += 256) fw_s[i] = fw[i];
    if (tid < CF) fb_s[tid] = fb[tid];
    __syncthreads();

    const int lane  = tid & 31;
    const int wv    = tid >> 5;
    const int m     = lane & 15;
    const int khalf = (lane < 16) ? 0 : 2;
    const float* xb = x + (size_t)b * CCH * HWSZ;

    for (int tt = 0; tt < 2; ++tt) {
        const int t    = wv * 2 + tt;          // n-tile 0..15
        const int node = t * 16 + m;
        const int sp   = (wgi * 16 + (node >> 4)) * HDIM + hgj * 16 + (node & 15);
        v8f acc = {};
        #pragma unroll
        for (int s = 0; s < 32; ++s) {         // K = 128, 4 per WMMA
            const int k0 = 4 * s + khalf;
            v2f a, bm;
            a.x  = fw_s[m * CCH + k0];
            a.y  = fw_s[m * CCH + k0 + 1];
            bm.x = xb[(size_t)k0 * HWSZ + sp];
            bm.y = xb[(size_t)(k0 + 1) * HWSZ + sp];
            acc = __builtin_amdgcn_wmma_f32_16x16x4_f32(
                false, a, false, bm, (short)0, acc, false, false);
        }
        #pragma unroll
        for (int r = 0; r < 8; ++r) {
            const int d = (lane < 16) ? r : (r + 8);
            ftile[node * CF + d] = acc[r] + fb_s[d];
        }
    }
    __syncthreads();

    // per-node: append normalized grid coords, L2-normalize (eps 1e-8)
    const int n = tid;
    float v[DAUG];
    #pragma unroll
    for (int j = 0; j < CF; ++j) v[j] = ftile[n * CF + j];
    v[16] = grid_norm(n >> 4);
    v[17] = grid_norm(n & 15);
    float ss = 0.f;
    #pragma unroll
    for (int j = 0; j < DAUG; ++j) ss += v[j] * v[j];
    const float inv = 1.0f / fmaxf(sqrtf(ss), 1e-8f);
    float* o = xn + ((size_t)p * NNODE + n) * DPAD;
    #pragma unroll
    for (int j = 0; j < DAUG; ++j) o[j] = v[j] * inv;
    o[18] = 0.f;
    o[19] = 0.f;
}

// ---------------------------------------------------------------------------
// Kernel 2: cosine similarity sim = xn * xn^T per patch via f32 WMMA (K=20)
// ---------------------------------------------------------------------------
__global__ void k2_sim(const float* __restrict__ xn, float* __restrict__ sim) {
    __shared__ float xs[NNODE * DPAD];   // 20 KB

    const int tid = threadIdx.x;
    const int p   = blockIdx.x;
    const float* src = xn + (size_t)p * NNODE * DPAD;
    for (int i = tid; i < NNODE * DPAD; i += 256) xs[i] = src[i];
    __syncthreads();

    const int lane  = tid & 31;
    const int wv    = tid >> 5;
    const int m     = lane & 15;
    const int khalf = (lane < 16) ? 0 : 2;
    float* sp = sim + (size_t)p * NNODE * NNODE;

    for (int t = 0; t < 32; ++t) {            // 256 tiles / 8 waves
        const int tile = t * 8 + wv;
        const int ti = tile >> 4, tj = tile & 15;
        v8f acc = {};
        #pragma unroll
        for (int s = 0; s < 5; ++s) {         // K = 20 (last 2 dims are zero pad)
            const int k0 = 4 * s + khalf;
            v2f a, bm;
            a.x  = xs[(ti * 16 + m) * DPAD + k0];
            a.y  = xs[(ti * 16 + m) * DPAD + k0 + 1];
            bm.x = xs[(tj * 16 + m) * DPAD + k0];
            bm.y = xs[(tj * 16 + m) * DPAD + k0 + 1];
            acc = __builtin_amdgcn_wmma_f32_16x16x4_f32(
                false, a, false, bm, (short)0, acc, false, false);
        }
        #pragma unroll
        for (int r = 0; r < 8; ++r) {
            const int mm = ti * 16 + ((lane < 16) ? r : (r + 8));
            sp[(size_t)mm * NNODE + tj * 16 + m] = acc[r];
        }
    }
}

// ---------------------------------------------------------------------------
// Kernel 3: per-row top-15 (wave32 butterfly argmax), sigmoid+softmax weights,
//           weighted channel gather -> un-patchified output
// ---------------------------------------------------------------------------
__global__ void k3_topk_agg(const float* __restrict__ x,
                            const float* __restrict__ sim,
                            const float* __restrict__ alpha,
                            const float* __restrict__ beta,
                            float* __restrict__ out) {
    const int tid  = threadIdx.x;
    const int lane = tid & 31;
    const int wv   = tid >> 5;
    const int row  = blockIdx.x * 8 + wv;     // p*256 + n
    const int p = row >> 8, n = row & 255;
    const int b = p / 49, pr = p % 49, wgi = pr / 7, hgj = pr % 7;

    const float* srow = sim + (size_t)row * NNODE;
    float vals[8];
    #pragma unroll
    for (int q = 0; q < 8; ++q) vals[q] = srow[lane + 32 * q];

    const float NEG = -__builtin_inff();
    float topv[KTOP];
    int   topi[KTOP];
    #pragma unroll
    for (int k = 0; k < KTOP; ++k) {
        float bv = NEG; int bi = 0x7fffffff;
        #pragma unroll
        for (int q = 0; q < 8; ++q) {
            const int ci = lane + 32 * q;
            if (vals[q] > bv || (vals[q] == bv && ci < bi)) { bv = vals[q]; bi = ci; }
        }
        #pragma unroll
        for (int off = 16; off > 0; off >>= 1) {
            const float ov = __shfl_xor(bv, off, 32);
            const int   oi = __shfl_xor(bi, off, 32);
            if (ov > bv || (ov == bv && oi < bi)) { bv = ov; bi = oi; }
        }
        topv[k] = bv; topi[k] = bi;
        const int qq = bi >> 5;
        #pragma unroll
        for (int q = 0; q < 8; ++q)
            if (q == qq && (bi & 31) == lane) vals[q] = NEG;
    }

    const float al = alpha[0], be = beta[0];
    float e[KTOP];
    float sum = 0.f;
    #pragma unroll
    for (int k = 0; k < KTOP; ++k) {
        const float wk = 1.0f / (1.0f + expf(-(be + al * topv[k]))); // sigmoid
        e[k] = expf(wk);                                             // softmax num
        sum += e[k];
    }
    const float rinv = 1.0f / sum;

    // each lane aggregates 4 channels: lane, lane+32, lane+64, lane+96
    const int spb = wgi * 16 * HDIM + hgj * 16;
    const float* xb = x + (size_t)b * CCH * HWSZ;
    float a0 = 0.f, a1 = 0.f, a2 = 0.f, a3 = 0.f;
    #pragma unroll
    for (int k = 0; k < KTOP; ++k) {
        const float wn = e[k] * rinv;
        const int node = topi[k];
        const int sp = spb + (node >> 4) * HDIM + (node & 15);
        a0 += wn * xb[(size_t)(lane     ) * HWSZ + sp];
        a1 += wn * xb[(size_t)(lane + 32) * HWSZ + sp];
        a2 += wn * xb[(size_t)(lane + 64) * HWSZ + sp];
        a3 += wn * xb[(size_t)(lane + 96) * HWSZ + sp];
    }

    const int spo = spb + (n >> 4) * HDIM + (n & 15);
    float* ob = out + (size_t)b * CCH * HWSZ;
    ob[(size_t)(lane     ) * HWSZ + spo] = a0;
    ob[(size_t)(lane + 32) * HWSZ + spo] = a1;
    ob[(size_t)(lane + 64) * HWSZ + spo] = a2;
    ob[(size_t)(lane + 96) * HWSZ + spo] = a3;
}

extern "C" void kernel_launch(void* const* d_in, const int* in_sizes, int n_in,
                              void* d_out, int out_size, void* d_ws, size_t ws_size,
                              hipStream_t stream) {
    const float* x     = (const float*)d_in[0];   // [4,128,112,112]
    const float* fw    = (const float*)d_in[1];   // [16,128]
    const float* fb    = (const float*)d_in[2];   // [16]
    const float* alpha = (const float*)d_in[3];   // [1]
    const float* beta  = (const float*)d_in[4];   // [1]
    float* out = (float*)d_out;

    // workspace layout: xn at 0 (196*256*20 f32 ~ 3.9 MB), sim at +4 MB (~49 MB)
    float* xn  = (float*)d_ws;
    float* sim = (float*)((char*)d_ws + ((size_t)4 << 20));

    k1_conv_feat<<<NPATCH, 256, 0, stream>>>(x, fw, fb, xn);
    k2_sim<<<NPATCH, 256, 0, stream>>>(xn, sim);
    k3_topk_agg<<<NPATCH * 32, 256, 0, stream>>>(x, sim, alpha, beta, out);
}